// LplsNorm_88768384074042
// MI455X (gfx1250) — compile-verified
//
#include <hip/hip_runtime.h>

#define N 8192
#define N4 (N / 4)

typedef float v2f __attribute__((ext_vector_type(2)));
typedef float v4f __attribute__((ext_vector_type(4)));
typedef float v8f __attribute__((ext_vector_type(8)));

// ---------------------------------------------------------------------------
// Kernel 1: row sums via V_WMMA_F32_16X16X4_F32.
// A-operand = all ones  =>  D[m][n] = sum_k B[k][n]  (independent of m).
// Addition is commutative, so we can feed 64 row elements (2 of every 4 in a 128-float span) per
// WMMA into the B operand with perfectly coalesced loads; the lane<->(k,n)
// mapping is irrelevant.  Horizontal sum of D[0][0..15] = chunk total.
// One wave per row; no divergence before the WMMAs (EXEC all ones).
// ---------------------------------------------------------------------------
__global__ __launch_bounds__(256) void lpls_degree_kernel(
    const float* __restrict__ A, float* __restrict__ deg) {
  const int lane = threadIdx.x & 31;
  const int wave = threadIdx.x >> 5;
  const int row  = blockIdx.x * 8 + wave;          // grid sized exactly: no guard
  const float* __restrict__ rp = A + (size_t)row * N;

  v2f ones;
  ones.x = 1.0f;
  ones.y = 1.0f;
  v8f acc = {};

  // 256 floats per wave-iteration: two coalesced b128 loads, four WMMAs.
  for (int c0 = 0; c0 < N; c0 += 256) {
    v4f q0 = *(const v4f*)(rp + c0 + lane * 4);          // 512B contiguous
    v4f q1 = *(const v4f*)(rp + c0 + 128 + lane * 4);    // next 512B
    v2f b0, b1, b2, b3;
    b0.x = q0.x; b0.y = q0.y;
    b1.x = q0.z; b1.y = q0.w;
    b2.x = q1.x; b2.y = q1.y;
    b3.x = q1.z; b3.y = q1.w;
    acc = __builtin_amdgcn_wmma_f32_16x16x4_f32(false, ones, false, b0,
                                                (short)0, acc, false, false);
    acc = __builtin_amdgcn_wmma_f32_16x16x4_f32(false, ones, false, b1,
                                                (short)0, acc, false, false);
    acc = __builtin_amdgcn_wmma_f32_16x16x4_f32(false, ones, false, b2,
                                                (short)0, acc, false, false);
    acc = __builtin_amdgcn_wmma_f32_16x16x4_f32(false, ones, false, b3,
                                                (short)0, acc, false, false);
  }

  // acc[0] lane n (n<16) holds D[0][n]; lanes 16..31 hold the identical
  // D[8][n].  Reduce within each 16-lane half; lane 0 holds the row sum.
  float s = acc[0];
  s += __shfl_xor(s, 1, 32);
  s += __shfl_xor(s, 2, 32);
  s += __shfl_xor(s, 4, 32);
  s += __shfl_xor(s, 8, 32);
  if (lane == 0) deg[row] = s;
}

// ---------------------------------------------------------------------------
// Kernel 2: deg -> deg^-1/2 in place (8192 elements, trivial).
// ---------------------------------------------------------------------------
__global__ __launch_bounds__(256) void lpls_invsqrt_kernel(float* __restrict__ d) {
  const int i = blockIdx.x * 256 + threadIdx.x;
  d[i] = 1.0f / sqrtf(d[i]);
}

// ---------------------------------------------------------------------------
// Kernel 3: out[r][c] = A[r][c] * inv[r] * inv[c], one float4 per thread.
// inv[r] is uniform per block (2048 float4 per row, 256-thread blocks);
// inv4[c] is a hot 32KB table.  A is on its last use and out is never
// re-read -> nontemporal on both sides of the streaming pass.
// ---------------------------------------------------------------------------
__global__ __launch_bounds__(256) void lpls_scale_kernel(
    const float* __restrict__ A, const float* __restrict__ inv,
    float* __restrict__ out) {
  const size_t gid = (size_t)blockIdx.x * 256 + threadIdx.x;  // < 2^24
  const int r  = (int)(gid >> 11);        // gid / (N/4)
  const int c4 = (int)(gid & (N4 - 1));   // gid % (N/4)

  v4f a  = __builtin_nontemporal_load((const v4f*)A + gid);
  v4f ic = *((const v4f*)inv + c4);
  const float sr = inv[r];

  v4f o;
  o.x = a.x * sr * ic.x;
  o.y = a.y * sr * ic.y;
  o.z = a.z * sr * ic.z;
  o.w = a.w * sr * ic.w;
  __builtin_nontemporal_store(o, (v4f*)out + gid);
}

extern "C" void kernel_launch(void* const* d_in, const int* in_sizes, int n_in,
                              void* d_out, int out_size, void* d_ws, size_t ws_size,
                              hipStream_t stream) {
  const float* A   = (const float*)d_in[0];
  float*       out = (float*)d_out;
  float*       deg = (float*)d_ws;   // 8192 floats = 32 KB scratch

  // Pass 1: row sums (reads 256 MB), one wave per row.
  lpls_degree_kernel<<<N / 8, 256, 0, stream>>>(A, deg);
  // Pass 2: deg -> deg^-1/2 (32 KB).
  lpls_invsqrt_kernel<<<N / 256, 256, 0, stream>>>(deg);
  // Pass 3: rescale (reads 256 MB + 32 KB, writes 256 MB).
  lpls_scale_kernel<<<(N * N4) / 256, 256, 0, stream>>>(A, deg, out);
}